// GCN_with_MLP_40415642256055
// MI455X (gfx1250) — compile-verified
//
#include <hip/hip_runtime.h>
#include <hip/hip_bf16.h>
#include <stdint.h>

#define BN_EPS_F 1e-5f

typedef __attribute__((ext_vector_type(2))) float v2f;
typedef __attribute__((ext_vector_type(8))) float v8f;

// row strides (in floats) for LDS tiles, padded for conflict-free fragment reads
#define XROW 132
#define WROW 136

static __device__ __forceinline__ void wait_async0() {
#if __has_builtin(__builtin_amdgcn_s_wait_asynccnt)
    __builtin_amdgcn_s_wait_asynccnt(0);
#else
    asm volatile("s_wait_asynccnt 0" ::: "memory");
#endif
}

// one per-lane 16B global->LDS async copy (gfx1250 GLOBAL_LOAD_ASYNC_TO_LDS_B128)
static __device__ __forceinline__ void async_copy_b128(uint32_t ldsOff, const float* g) {
    asm volatile("global_load_async_to_lds_b128 %0, %1, off" ::"v"(ldsOff), "v"(g) : "memory");
}

// ---------------------------------------------------------------- fill zero
__global__ __launch_bounds__(256) void fill_zero_kernel(float* __restrict__ p, long n) {
    long i = (long)blockIdx.x * blockDim.x + threadIdx.x;
    long stride = (long)gridDim.x * blockDim.x;
    for (; i < n; i += stride) p[i] = 0.0f;
}

// ---------------------------------------------------------------- degrees
__global__ __launch_bounds__(256) void degree_kernel(const int* __restrict__ src,
                                                     const int* __restrict__ dst,
                                                     float* __restrict__ out_deg,
                                                     float* __restrict__ in_deg, int nE) {
    int i = blockIdx.x * blockDim.x + threadIdx.x;
    int stride = gridDim.x * blockDim.x;
    for (; i < nE; i += stride) {
        unsafeAtomicAdd(&out_deg[src[i]], 1.0f);
        unsafeAtomicAdd(&in_deg[dst[i]], 1.0f);
    }
}

__global__ __launch_bounds__(256) void norm_kernel(const float* __restrict__ out_deg,
                                                   const float* __restrict__ in_deg,
                                                   float* __restrict__ norm_src,
                                                   float* __restrict__ norm_dst, int n) {
    int i = blockIdx.x * blockDim.x + threadIdx.x;
    if (i < n) {
        norm_src[i] = rsqrtf(fmaxf(out_deg[i], 1.0f));
        norm_dst[i] = rsqrtf(fmaxf(in_deg[i], 1.0f));
    }
}

// ---------------------------------------------------------------- BN column stats
// stats[0..127] = column sums, stats[128..255] = column sum-of-squares
__global__ __launch_bounds__(256) void bn_stats_kernel(const float* __restrict__ h,
                                                       float* __restrict__ stats, int nrows) {
    __shared__ float s1[256];
    __shared__ float s2[256];
    const int c = threadIdx.x & 127;
    const int half = threadIdx.x >> 7;
    const int rowBase = blockIdx.x * 128;
    float sum = 0.0f, sumsq = 0.0f;
    for (int r = half; r < 128; r += 2) {
        int gr = rowBase + r;
        if (gr < nrows) {
            float v = h[(size_t)gr * 128 + c];
            sum += v;
            sumsq += v * v;
        }
    }
    s1[threadIdx.x] = sum;
    s2[threadIdx.x] = sumsq;
    __syncthreads();
    if (half == 0) {
        unsafeAtomicAdd(&stats[c], s1[threadIdx.x] + s1[threadIdx.x + 128]);
        unsafeAtomicAdd(&stats[128 + c], s2[threadIdx.x] + s2[threadIdx.x + 128]);
    }
}

// ---------------------------------------------------------------- edge scatter
// agg[dst[e]] += relu(h[src[e]]) * norm_src[src[e]]   (one wave per edge, lane-coalesced)
__global__ __launch_bounds__(256) void scatter_kernel(const float* __restrict__ h,
                                                      const float* __restrict__ norm_src,
                                                      const int* __restrict__ src,
                                                      const int* __restrict__ dst,
                                                      float* __restrict__ agg, int nEdges) {
    const int gtid = blockIdx.x * blockDim.x + threadIdx.x;
    const int wid = gtid >> 5;
    const int lane = threadIdx.x & 31;
    const int nWaves = (gridDim.x * blockDim.x) >> 5;
    for (int e = wid; e < nEdges; e += nWaves) {
        const int s = src[e];
        const int d = dst[e];
        const float ns = norm_src[s];
        const float* __restrict__ hs = h + (size_t)s * 128;
        float* __restrict__ ad = agg + (size_t)d * 128;
#pragma unroll
        for (int q = 0; q < 4; ++q) {
            const int f = lane + 32 * q;
            float v = fmaxf(hs[f], 0.0f) * ns;
            unsafeAtomicAdd(ad + f, v);
        }
    }
}

// ---------------------------------------------------------------- WMMA fp32 GEMM
// Y[nrows,128] = X[nrows,128] @ W[128,128] (+ epilogue per MODE)
// MODE 0: Y = XW + bias
// MODE 1: Y = bnrelu(X) W + bias   (BN from stats colsum/colsumsq, gamma, beta)
// MODE 2: Y = (XW) * row_scale[row] + bias; if zero_after != null, re-zero the
//         consumed X slab (X == zero_after == agg) for the next propagation step.
template <int MODE>
__global__ __launch_bounds__(256) void gemm128_wmma_kernel(
    const float* X, const float* __restrict__ W,
    const float* __restrict__ bias, float* __restrict__ Y, int nrows, int n_total,
    const float* __restrict__ stats, const float* __restrict__ gamma,
    const float* __restrict__ beta, const float* __restrict__ row_scale,
    float* zero_after) {
    __shared__ float Ws[128][WROW];  // row-major: Ws[k][n] = W[k][n]
    __shared__ float Xs[128][XROW];  // row-major tile
    __shared__ float bnScale[128];
    __shared__ float bnShift[128];

    const int tid = threadIdx.x;
    const int rowBase = blockIdx.x * 128;

    if (MODE == 1 && tid < 128) {
        const float inv_n = 1.0f / (float)n_total;
        float mu = stats[tid] * inv_n;
        float var = stats[128 + tid] * inv_n - mu * mu;
        float is = rsqrtf(var + BN_EPS_F);
        bnScale[tid] = gamma[tid] * is;
        bnShift[tid] = beta[tid] - mu * gamma[tid] * is;
    }

    // async-stage W (row-major, 16B per lane per issue)
    {
        const uint32_t ldsW = (uint32_t)(uintptr_t)&Ws[0][0];
        for (int i = tid * 4; i < 128 * 128; i += 1024) {
            const int k = i >> 7;
            const int n = i & 127;
            async_copy_b128(ldsW + (uint32_t)(k * WROW + n) * 4u, W + i);
        }
    }
    // async-stage X tile; zero-fill out-of-range rows
    {
        const uint32_t ldsX = (uint32_t)(uintptr_t)&Xs[0][0];
        for (int i = tid * 4; i < 128 * 128; i += 1024) {
            const int r = i >> 7;
            const int k = i & 127;
            const int gr = rowBase + r;
            if (gr < nrows) {
                async_copy_b128(ldsX + (uint32_t)(r * XROW + k) * 4u,
                                X + (size_t)gr * 128 + k);
            } else {
                *(float4*)&Xs[r][k] = make_float4(0.0f, 0.0f, 0.0f, 0.0f);
            }
        }
    }
    wait_async0();
    __syncthreads();

    // fused BN + ReLU applied in-place in LDS
    if (MODE == 1) {
        for (int i = tid; i < 128 * 128; i += 256) {
            const int r = i >> 7;
            const int k = i & 127;
            Xs[r][k] = fmaxf(fmaf(Xs[r][k], bnScale[k], bnShift[k]), 0.0f);
        }
        __syncthreads();
    }

    // re-zero the consumed agg slab for the next propagation step
    if (MODE == 2 && zero_after != nullptr) {
        for (int i = tid * 4; i < 128 * 128; i += 1024) {
            const int r = i >> 7;
            const int k = i & 127;
            const int gr = rowBase + r;
            if (gr < nrows)
                *(float4*)(zero_after + (size_t)gr * 128 + k) =
                    make_float4(0.0f, 0.0f, 0.0f, 0.0f);
        }
    }

    // wave tiling: 8 waves -> 4 row-groups (32 rows) x 2 col-groups (64 cols)
    const int wave = tid >> 5;
    const int mi = wave & 3;
    const int ni = wave >> 2;
    const int lane = tid & 31;
    const int halfSel = lane >> 4;  // K-pair select (A/B), M-half select (C)
    const int mN = lane & 15;       // M for A, N for B/C

    v8f acc[2][4] = {};

    for (int kb = 0; kb < 32; ++kb) {
        const int k0 = kb * 4 + halfSel * 2;
        v2f a0, a1;
        a0.x = Xs[mi * 32 + mN][k0];
        a0.y = Xs[mi * 32 + mN][k0 + 1];
        a1.x = Xs[mi * 32 + 16 + mN][k0];
        a1.y = Xs[mi * 32 + 16 + mN][k0 + 1];
#pragma unroll
        for (int nt = 0; nt < 4; ++nt) {
            const int col = ni * 64 + nt * 16 + mN;
            v2f b;
            b.x = Ws[k0][col];
            b.y = Ws[k0 + 1][col];
            acc[0][nt] = __builtin_amdgcn_wmma_f32_16x16x4_f32(
                false, a0, false, b, (short)0, acc[0][nt], false, false);
            acc[1][nt] = __builtin_amdgcn_wmma_f32_16x16x4_f32(
                false, a1, false, b, (short)0, acc[1][nt], false, false);
        }
    }

    // C layout: VGPR j -> M = j + 8*halfSel, N = lane%16
#pragma unroll
    for (int mt = 0; mt < 2; ++mt) {
        const int rowHalf = rowBase + mi * 32 + mt * 16 + 8 * halfSel;
#pragma unroll
        for (int j = 0; j < 8; ++j) {
            const int r = rowHalf + j;
            if (r < nrows) {
                const float rs = (MODE == 2) ? row_scale[r] : 1.0f;
#pragma unroll
                for (int nt = 0; nt < 4; ++nt) {
                    const int col = ni * 64 + nt * 16 + mN;
                    float v = acc[mt][nt][j];
                    if (MODE == 2)
                        v = v * rs + bias[col];
                    else
                        v = v + bias[col];
                    Y[(size_t)r * 128 + col] = v;
                }
            }
        }
    }
}

// ---------------------------------------------------------------- launch
extern "C" void kernel_launch(void* const* d_in, const int* in_sizes, int n_in,
                              void* d_out, int out_size, void* d_ws, size_t ws_size,
                              hipStream_t stream) {
    const float* in_feat = (const float*)d_in[0];
    const int* src       = (const int*)d_in[1];
    const int* dst       = (const int*)d_in[2];
    const float* W1      = (const float*)d_in[3];
    const float* b1      = (const float*)d_in[4];
    const float* gamma   = (const float*)d_in[5];
    const float* beta    = (const float*)d_in[6];
    const float* W2      = (const float*)d_in[7];
    const float* b2      = (const float*)d_in[8];
    const float* Wc      = (const float*)d_in[9];
    const float* bc      = (const float*)d_in[10];
    float* out = (float*)d_out;

    const int N = in_sizes[0] / 128;  // 100000 nodes
    const int E = in_sizes[1];        // 1600000 edges

    // workspace partition (floats)
    float* ws = (float*)d_ws;
    size_t off = 0;
    float* stats    = ws + off; off += 256;
    float* out_deg  = ws + off; off += (size_t)N;
    float* in_deg   = ws + off; off += (size_t)N;   // contiguous with out_deg
    float* norm_src = ws + off; off += (size_t)N;
    float* norm_dst = ws + off; off += (size_t)N;
    float* hA       = ws + off; off += (size_t)N * 128;
    float* hB       = ws + off; off += (size_t)N * 128;
    float* agg      = ws + off; off += (size_t)N * 128;

    const int gemmBlocks = (N + 127) / 128;

    // zero stats + both degree arrays
    fill_zero_kernel<<<1, 256, 0, stream>>>(stats, 256);
    fill_zero_kernel<<<512, 256, 0, stream>>>(out_deg, 2L * N);

    // MLP layer 1: hA = in_feat @ W1 + b1
    gemm128_wmma_kernel<0><<<gemmBlocks, 256, 0, stream>>>(
        in_feat, W1, b1, hA, N, N, nullptr, nullptr, nullptr, nullptr, nullptr);

    // BN batch statistics over hA
    bn_stats_kernel<<<gemmBlocks, 256, 0, stream>>>(hA, stats, N);

    // degrees + symmetric norms
    degree_kernel<<<1024, 256, 0, stream>>>(src, dst, out_deg, in_deg, E);
    norm_kernel<<<(N + 255) / 256, 256, 0, stream>>>(out_deg, in_deg, norm_src, norm_dst, N);

    // MLP layer 2 with fused BN+ReLU on input: hB = relu(bn(hA)) @ W2 + b2
    gemm128_wmma_kernel<1><<<gemmBlocks, 256, 0, stream>>>(
        hA, W2, b2, hB, N, N, stats, gamma, beta, nullptr, nullptr);

    // initial zero of agg (later steps re-zero inside the MODE-2 GEMM epilogue)
    fill_zero_kernel<<<2048, 256, 0, stream>>>(agg, (long)N * 128);

    // 3 propagation steps: agg = scatter(relu(h)*norm_src); h = (agg@Wc)*norm_dst + bc
    float* cur = hB;
    float* nxt = hA;
    for (int step = 0; step < 3; ++step) {
        scatter_kernel<<<4096, 256, 0, stream>>>(cur, norm_src, src, dst, agg, E);
        float* outBuf = (step == 2) ? out : nxt;
        float* zeroPtr = (step < 2) ? agg : nullptr;  // last step: no need to re-zero
        gemm128_wmma_kernel<2><<<gemmBlocks, 256, 0, stream>>>(
            agg, Wc, bc, outBuf, N, N, nullptr, nullptr, nullptr, norm_dst, zeroPtr);
        float* t = cur; cur = nxt; nxt = t;
    }
}